// GCN_90305982365931
// MI455X (gfx1250) — compile-verified
//
#include <hip/hip_runtime.h>
#include <stdint.h>

typedef __attribute__((ext_vector_type(2))) float v2f;
typedef __attribute__((ext_vector_type(8))) float v8f;

// ---------------------------------------------------------------------------
// Degree / norm kernels
// ---------------------------------------------------------------------------
__global__ __launch_bounds__(256) void deg_init(float* __restrict__ deg, int N) {
    unsigned i = blockIdx.x * 256u + threadIdx.x;
    if (i < (unsigned)N) deg[i] = 1.0f;  // self-loop contributes 1 to every target degree
}

__global__ __launch_bounds__(256) void deg_count(const long long* __restrict__ col,
                                                 float* __restrict__ deg, int E) {
    unsigned e = blockIdx.x * 256u + threadIdx.x;
    if (e < (unsigned)E) unsafeAtomicAdd(&deg[(int)col[e]], 1.0f);
}

__global__ __launch_bounds__(256) void deg_to_dinv(float* __restrict__ deg, int N) {
    unsigned i = blockIdx.x * 256u + threadIdx.x;
    if (i < (unsigned)N) deg[i] = rsqrtf(deg[i]);  // deg >= 1 always
}

__global__ __launch_bounds__(256) void edge_norm(const long long* __restrict__ row,
                                                 const long long* __restrict__ col,
                                                 const float* __restrict__ dinv,
                                                 float* __restrict__ nrm, int E) {
    unsigned e = blockIdx.x * 256u + threadIdx.x;
    if (e < (unsigned)E) nrm[e] = dinv[(int)row[e]] * dinv[(int)col[e]];
}

// ---------------------------------------------------------------------------
// GEMM1: H1[M x 128] = X[M x 256] @ W1[256 x 128], fp32 WMMA 16x16x4.
// One wave per 16x16 tile; 8 waves/block cover the full N=128 width.
// Two independent accumulator chains; branch-free full-tile store path.
// ---------------------------------------------------------------------------
__global__ __launch_bounds__(256) void gemm1_wmma(const float* __restrict__ X,
                                                  const float* __restrict__ W,
                                                  float* __restrict__ H, int M) {
    const int wave = threadIdx.x >> 5;       // 0..7 -> N tile
    const int lane = threadIdx.x & 31;
    const int hi   = lane >> 4;              // half of wave
    const int lm   = lane & 15;
    const int m0   = blockIdx.x * 16;
    const int n0   = wave * 16;
    if (m0 >= M) return;                     // wave-uniform

    const int mr = (m0 + lm < M) ? (m0 + lm) : (M - 1);  // clamp (EXEC stays full)
    const float* __restrict__ arow = X + (size_t)mr * 256;

    v8f c0 = {}, c1 = {};
#pragma unroll 4
    for (int kk = 0; kk < 256; kk += 8) {
        const int k0 = kk + 2 * hi;          // A/B VGPR0 holds K=2*hi, VGPR1 holds K=2*hi+1
        v2f a0; a0.x = arow[k0];             a0.y = arow[k0 + 1];
        v2f b0; b0.x = W[(size_t)k0 * 128 + n0 + lm];
                b0.y = W[(size_t)(k0 + 1) * 128 + n0 + lm];
        c0 = __builtin_amdgcn_wmma_f32_16x16x4_f32(false, a0, false, b0,
                                                   (short)0, c0, false, false);
        const int k1 = kk + 4 + 2 * hi;
        v2f a1; a1.x = arow[k1];             a1.y = arow[k1 + 1];
        v2f b1; b1.x = W[(size_t)k1 * 128 + n0 + lm];
                b1.y = W[(size_t)(k1 + 1) * 128 + n0 + lm];
        c1 = __builtin_amdgcn_wmma_f32_16x16x4_f32(false, a1, false, b1,
                                                   (short)0, c1, false, false);
    }

    float* __restrict__ hbase = H + (size_t)(m0 + 8 * hi) * 128 + n0 + lm;
    if (m0 + 16 <= M) {                      // wave-uniform fast path: no per-store guards
#pragma unroll
        for (int v = 0; v < 8; ++v) hbase[(size_t)v * 128] = c0[v] + c1[v];
    } else {
#pragma unroll
        for (int v = 0; v < 8; ++v)
            if (m0 + v + 8 * hi < M) hbase[(size_t)v * 128] = c0[v] + c1[v];
    }
}

// ---------------------------------------------------------------------------
// GEMM2: H2[M x 16] = relu(AGG1)[M x 128] @ W2[128 x 16], ReLU fused on A load.
// 8 waves/block -> 8 M-tiles per block (N=16 is a single tile).
// ---------------------------------------------------------------------------
__global__ __launch_bounds__(256) void gemm2_wmma(const float* __restrict__ AGG,
                                                  const float* __restrict__ W,
                                                  float* __restrict__ H, int M) {
    const int wave = threadIdx.x >> 5;
    const int lane = threadIdx.x & 31;
    const int hi   = lane >> 4;
    const int lm   = lane & 15;
    const int m0   = (blockIdx.x * 8 + wave) * 16;
    if (m0 >= M) return;                     // wave-uniform

    const int mr = (m0 + lm < M) ? (m0 + lm) : (M - 1);
    const float* __restrict__ arow = AGG + (size_t)mr * 128;

    v8f c0 = {}, c1 = {};
#pragma unroll 4
    for (int kk = 0; kk < 128; kk += 8) {
        const int k0 = kk + 2 * hi;
        v2f a0; a0.x = fmaxf(arow[k0], 0.0f);     a0.y = fmaxf(arow[k0 + 1], 0.0f);
        v2f b0; b0.x = W[(size_t)k0 * 16 + lm];
                b0.y = W[(size_t)(k0 + 1) * 16 + lm];
        c0 = __builtin_amdgcn_wmma_f32_16x16x4_f32(false, a0, false, b0,
                                                   (short)0, c0, false, false);
        const int k1 = kk + 4 + 2 * hi;
        v2f a1; a1.x = fmaxf(arow[k1], 0.0f);     a1.y = fmaxf(arow[k1 + 1], 0.0f);
        v2f b1; b1.x = W[(size_t)k1 * 16 + lm];
                b1.y = W[(size_t)(k1 + 1) * 16 + lm];
        c1 = __builtin_amdgcn_wmma_f32_16x16x4_f32(false, a1, false, b1,
                                                   (short)0, c1, false, false);
    }

    float* __restrict__ hbase = H + (size_t)(m0 + 8 * hi) * 16 + lm;
    if (m0 + 16 <= M) {
#pragma unroll
        for (int v = 0; v < 8; ++v) hbase[(size_t)v * 16] = c0[v] + c1[v];
    } else {
#pragma unroll
        for (int v = 0; v < 8; ++v)
            if (m0 + v + 8 * hi < M) hbase[(size_t)v * 16] = c0[v] + c1[v];
    }
}

// ---------------------------------------------------------------------------
// agg[n][d..d+3] = dinv[n]^2 * h[n][d..d+3] + bias[d..d+3]   (float4, overwrite)
// ---------------------------------------------------------------------------
template <int D>
__global__ __launch_bounds__(256) void self_init(const float* __restrict__ dinv,
                                                 const float* __restrict__ h,
                                                 const float* __restrict__ bias,
                                                 float* __restrict__ agg, int N) {
    constexpr int Q = D / 4;                 // float4 groups per node
    unsigned gid = blockIdx.x * 256u + threadIdx.x;
    if (gid >= (unsigned)N * Q) return;
    unsigned n = gid / Q, q = gid % Q;
    float di = dinv[n];
    float s  = di * di;
    float4 hv = ((const float4*)h)[(size_t)n * Q + q];
    float4 bv = ((const float4*)bias)[q];
    float4 o;
    o.x = s * hv.x + bv.x;  o.y = s * hv.y + bv.y;
    o.z = s * hv.z + bv.z;  o.w = s * hv.w + bv.w;
    ((float4*)agg)[(size_t)n * Q + q] = o;
}

// ---------------------------------------------------------------------------
// agg[col[e]][d..d+3] += norm[e] * h[row[e]][d..d+3]   (D/4 threads per edge)
// ---------------------------------------------------------------------------
template <int D>
__global__ __launch_bounds__(256) void edge_scatter(const long long* __restrict__ row,
                                                    const long long* __restrict__ col,
                                                    const float* __restrict__ nrm,
                                                    const float* __restrict__ h,
                                                    float* __restrict__ agg, int E) {
    constexpr int Q = D / 4;
    unsigned gid = blockIdx.x * 256u + threadIdx.x;
    unsigned e = gid / Q, q = gid % Q;
    if (e >= (unsigned)E) return;
    int r = (int)row[e], c = (int)col[e];
    float w = nrm[e];
    float4 hv = ((const float4*)h)[(size_t)r * Q + q];        // global_load_b128
    float* dst = agg + (size_t)c * D + q * 4;
    unsafeAtomicAdd(dst + 0, w * hv.x);
    unsafeAtomicAdd(dst + 1, w * hv.y);
    unsafeAtomicAdd(dst + 2, w * hv.z);
    unsafeAtomicAdd(dst + 3, w * hv.w);
}

// ---------------------------------------------------------------------------
extern "C" void kernel_launch(void* const* d_in, const int* in_sizes, int n_in,
                              void* d_out, int out_size, void* d_ws, size_t ws_size,
                              hipStream_t stream) {
    const float*     x   = (const float*)d_in[0];
    const long long* ei  = (const long long*)d_in[1];   // int64 edge_index [2, E]
    const float*     W1  = (const float*)d_in[2];
    const float*     b1  = (const float*)d_in[3];
    const float*     W2  = (const float*)d_in[4];
    const float*     b2  = (const float*)d_in[5];

    const int N = in_sizes[0] / 256;   // 100000 nodes
    const int E = in_sizes[1] / 2;     // 1600000 edges
    const long long* rowi = ei;        // sources
    const long long* coli = ei + E;    // targets

    // Workspace layout (256B aligned): dinv[N], norm[E], h1[N*128] (reused as h2), agg1[N*128]
    char*  ws  = (char*)d_ws;
    size_t off = 0;
    auto carve = [&](size_t bytes) {
        size_t o = off;
        off += (bytes + 255) & ~(size_t)255;
        return o;
    };
    float* dinv = (float*)(ws + carve((size_t)N * 4));
    float* nrm  = (float*)(ws + carve((size_t)E * 4));
    float* h1   = (float*)(ws + carve((size_t)N * 128 * 4));
    float* agg1 = (float*)(ws + carve((size_t)N * 128 * 4));
    float* h2   = h1;                  // h1 is dead after scatter1; stream order guarantees safety
    float* out  = (float*)d_out;

    auto blocks = [](size_t threads) { return (unsigned)((threads + 255) / 256); };

    // --- normalization ---
    deg_init   <<<blocks(N), 256, 0, stream>>>(dinv, N);
    deg_count  <<<blocks(E), 256, 0, stream>>>(coli, dinv, E);
    deg_to_dinv<<<blocks(N), 256, 0, stream>>>(dinv, N);
    edge_norm  <<<blocks(E), 256, 0, stream>>>(rowi, coli, dinv, nrm, E);

    // --- layer 1 ---
    gemm1_wmma<<<(N + 15) / 16, 256, 0, stream>>>(x, W1, h1, N);
    self_init<128><<<blocks((size_t)N * 32), 256, 0, stream>>>(dinv, h1, b1, agg1, N);
    edge_scatter<128><<<blocks((size_t)E * 32), 256, 0, stream>>>(rowi, coli, nrm, h1, agg1, E);

    // --- layer 2 (ReLU fused into GEMM2 A-operand) ---
    gemm2_wmma<<<(N / 16 + 7) / 8, 256, 0, stream>>>(agg1, W2, h2, N);
    self_init<16><<<blocks((size_t)N * 4), 256, 0, stream>>>(dinv, h2, b2, out, N);
    edge_scatter<16><<<blocks((size_t)E * 4), 256, 0, stream>>>(rowi, coli, nrm, h2, out, E);
}